// Q_Act_65481071395094
// MI455X (gfx1250) — compile-verified
//
#include <hip/hip_runtime.h>
#include <stdint.h>

#define B_    4
#define L_    4096
#define D_    2048
#define R_    10
#define G_    410
#define NROW  (B_ * G_)          // 1640
#define TPB   256
#define NWAVE (TPB / 32)         // 8 wave32 waves
#define EPT   (D_ / TPB)         // 8
#define QMAXF   7.0f
#define OQMAXF  127.0f

typedef uint32_t u32x4 __attribute__((ext_vector_type(4)));
typedef int32_t  i32x8 __attribute__((ext_vector_type(8)));
typedef int32_t  i32x4 __attribute__((ext_vector_type(4)));

// ---- wave32 butterfly + one LDS round: reduce (sum, sumsq, max) across the block ----
// lds must hold 3*NWAVE floats. Two barriers total.
__device__ __forceinline__ void block_reduce3(float& sum, float& ss, float& mx, float* lds)
{
#pragma unroll
    for (int off = 16; off > 0; off >>= 1) {
        sum += __shfl_xor(sum, off, 32);
        ss  += __shfl_xor(ss,  off, 32);
        mx   = fmaxf(mx, __shfl_xor(mx, off, 32));
    }
    const int wid  = threadIdx.x >> 5;
    const int lane = threadIdx.x & 31;
    if (lane == 0) { lds[wid] = sum; lds[NWAVE + wid] = ss; lds[2 * NWAVE + wid] = mx; }
    __syncthreads();
    float s0 = 0.0f, q0 = 0.0f, m0 = 0.0f;
#pragma unroll
    for (int i = 0; i < NWAVE; ++i) {
        s0 += lds[i]; q0 += lds[NWAVE + i]; m0 = fmaxf(m0, lds[2 * NWAVE + i]);
    }
    __syncthreads();                 // lds reusable after this
    sum = s0; ss = q0; mx = m0;
}

// max-only block reduction (two barriers)
__device__ __forceinline__ float block_reduce_max(float mx, float* lds)
{
#pragma unroll
    for (int off = 16; off > 0; off >>= 1) mx = fmaxf(mx, __shfl_xor(mx, off, 32));
    const int wid  = threadIdx.x >> 5;
    const int lane = threadIdx.x & 31;
    if (lane == 0) lds[wid] = mx;
    __syncthreads();
    float m0 = 0.0f;
#pragma unroll
    for (int i = 0; i < NWAVE; ++i) m0 = fmaxf(m0, lds[i]);
    __syncthreads();
    return m0;
}

// ---------------- step 0 stats + mask zero-init ----------------
__global__ void qact_init(const float* __restrict__ x, const float* __restrict__ s,
                          unsigned char* __restrict__ mask, float2* __restrict__ stats,
                          unsigned* __restrict__ flags)
{
    __shared__ float red[3 * NWAVE];
    const int row = blockIdx.x;
    const int b = row / G_, g = row % G_;
    const int l = g * R_;                          // always < L_
    const int tid = threadIdx.x;
    const int d0 = tid * EPT;
    const size_t xoff = ((size_t)b * L_ + l) * D_ + d0;

    // deterministic per-call mask reset (8 bytes/thread)
    *(uint2*)(mask + (size_t)row * D_ + d0) = make_uint2(0u, 0u);

    float4 xa = *(const float4*)(x + xoff);
    float4 xb = *(const float4*)(x + xoff + 4);
    float v[EPT] = {xa.x, xa.y, xa.z, xa.w, xb.x, xb.y, xb.z, xb.w};

    float sum = 0.0f, ss = 0.0f, mx = 0.0f;
#pragma unroll
    for (int i = 0; i < EPT; ++i) { float a = fabsf(v[i]); sum += a; ss += a * a; mx = fmaxf(mx, a); }

    block_reduce3(sum, ss, mx, red);
    if (tid == 0) {
        stats[row] = make_float2(sum, ss);         // slot r=0
        if (mx / QMAXF > s[l]) flags[0] = 1u;      // benign race: all write 1
    }
}

// ---------------- fused step r: quantize token l, prefetch+stats for token l+1 ----------------
__global__ void qact_step(const float* __restrict__ x, const float* __restrict__ s,
                          float* __restrict__ y,
                          unsigned char* __restrict__ mask, float2* __restrict__ stats,
                          unsigned* __restrict__ flags, int r)
{
    __shared__ float red[3 * NWAVE];
    __shared__ float xnext[D_];                    // 8KB TDM landing zone
    const int row = blockIdx.x;
    const int b = row / G_, g = row % G_;
    const int l = g * R_ + r;
    if (l >= L_) return;                           // padded token: no effect, output discarded
    const int tid = threadIdx.x;
    const int d0 = tid * EPT;
    const bool haveNext = (r + 1 < R_) && (l + 1 < L_);

    // --- kick off TDM DMA of next token's row into LDS (overlaps with main work) ---
    if (haveNext && tid == 0) {
        const float* xn = x + ((size_t)b * L_ + (l + 1)) * D_;
        uint64_t ga = (uint64_t)(uintptr_t)xn;
        uint32_t lo = (uint32_t)(uintptr_t)(&xnext[0]);   // flat->LDS offset (addr[31:0])
        u32x4 g0;
        g0[0] = 1u;                                            // count=1, user descriptor
        g0[1] = lo;                                            // lds_addr
        g0[2] = (uint32_t)ga;                                  // global_addr[31:0]
        g0[3] = (uint32_t)((ga >> 32) & 0x01FFFFFFu) | (2u << 30); // addr[56:32] | type=2
        i32x8 g1;
        g1[0] = (int)(2u << 16);                  // data_size=4B, no multicast/pad/iterate
        g1[1] = (int)((uint32_t)D_ << 16);        // tensor_dim0[15:0]=2048
        g1[2] = (int)(1u << 16);                  // tensor_dim0 hi=0 | tensor_dim1=1
        g1[3] = (int)((uint32_t)D_ << 16);        // tensor_dim1 hi=0 | tile_dim0=2048
        g1[4] = 0;                                // tile_dim1=0 (unused), tile_dim2=0
        g1[5] = (int)D_;                          // tensor_dim0_stride lo
        g1[6] = (int)((uint32_t)D_ << 16);        // stride0 hi=0 | tensor_dim1_stride lo
        g1[7] = 0;                                // stride1 hi
        i32x4 gz4 = {0, 0, 0, 0};                 // groups 2/3 unused (<=2D tensor)
        i32x8 gz8 = {0, 0, 0, 0, 0, 0, 0, 0};
        __builtin_amdgcn_tensor_load_to_lds(g0, g1, gz4, gz4, gz8, 0);
    }

    // --- main step-r work ---
    const size_t xoff = ((size_t)b * L_ + l) * D_ + d0;
    float4 xa = *(const float4*)(x + xoff);
    float4 xb = *(const float4*)(x + xoff + 4);
    float v[EPT] = {xa.x, xa.y, xa.z, xa.w, xb.x, xb.y, xb.z, xb.w};

    uint2 mw = *(const uint2*)(mask + (size_t)row * D_ + d0);
    float m[EPT];
#pragma unroll
    for (int i = 0; i < 4; ++i) m[i]     = ((mw.x >> (8 * i)) & 0xFFu) ? 1.0f : 0.0f;
#pragma unroll
    for (int i = 0; i < 4; ++i) m[i + 4] = ((mw.y >> (8 * i)) & 0xFFu) ? 1.0f : 0.0f;

    const unsigned fl = flags[r];                  // global any(dyn_s > s) branch
    float mn[EPT];
    if (fl) {
        float2 st = stats[r * NROW + row];         // sum, sumsq of |inlier| (pre-update mask)
        float mean = st.x * (1.0f / (float)D_);
        float var  = (st.y - st.x * st.x * (1.0f / (float)D_)) * (1.0f / (float)(D_ - 1));
        var = fmaxf(var, 0.0f);
        float thr = mean + 3.0f * sqrtf(var);
        uint32_t w0 = 0, w1 = 0;
#pragma unroll
        for (int i = 0; i < EPT; ++i) {
            float inl = v[i] * (1.0f - m[i]);
            float nm  = (m[i] > 0.0f || fabsf(inl) > thr) ? 1.0f : 0.0f;
            mn[i] = nm;
            uint32_t bs = (nm != 0.0f) ? 1u : 0u;
            if (i < 4) w0 |= bs << (8 * i); else w1 |= bs << (8 * (i - 4));
        }
        *(uint2*)(mask + (size_t)row * D_ + d0) = make_uint2(w0, w1);
    } else {
#pragma unroll
        for (int i = 0; i < EPT; ++i) mn[i] = m[i];
    }

    const float sE = s[l];
    float inx[EPT], oraw[EPT];
    float omx = 0.0f;
#pragma unroll
    for (int i = 0; i < EPT; ++i) {
        float inl2 = v[i] * (1.0f - mn[i]);
        float q = rintf(inl2 / sE);                         // round-half-even, like jnp.round
        q = fminf(fmaxf(q, -QMAXF), QMAXF);
        inx[i] = q * sE;
        float orw = v[i] * mn[i];
        oraw[i] = orw;
        omx = fmaxf(omx, fabsf(orw));
    }

    // per-row max|outlier| for the dynamic 8-bit scale
    const float omax = block_reduce_max(omx, red);
    const float out_s = fmaxf(omax, 1e-8f) / OQMAXF;

    float yv[EPT];
#pragma unroll
    for (int i = 0; i < EPT; ++i) {
        float q = rintf(oraw[i] / out_s);
        q = fminf(fmaxf(q, -OQMAXF), OQMAXF);
        yv[i] = inx[i] + q * out_s;
    }
    *(float4*)(y + xoff)     = make_float4(yv[0], yv[1], yv[2], yv[3]);
    *(float4*)(y + xoff + 4) = make_float4(yv[4], yv[5], yv[6], yv[7]);

    // --- next-step stats from the TDM-staged row (uses the just-updated mask) ---
    if (haveNext) {
        __builtin_amdgcn_s_wait_tensorcnt(0);      // wave0 waits on its DMA; others no-op
        __syncthreads();                           // publish LDS to all waves
        float4 na = *(const float4*)(&xnext[d0]);
        float4 nb = *(const float4*)(&xnext[d0 + 4]);
        float nv[EPT] = {na.x, na.y, na.z, na.w, nb.x, nb.y, nb.z, nb.w};
        float sum = 0.0f, ss = 0.0f, mx = 0.0f;
#pragma unroll
        for (int i = 0; i < EPT; ++i) {
            float a = fabsf(nv[i] * (1.0f - mn[i]));
            sum += a; ss += a * a; mx = fmaxf(mx, a);
        }
        block_reduce3(sum, ss, mx, red);
        if (tid == 0) {
            stats[(r + 1) * NROW + row] = make_float2(sum, ss);
            if (mx / QMAXF > s[l + 1]) flags[r + 1] = 1u;
        }
    }
}

// ---------------- host launcher (graph-capture safe) ----------------
extern "C" void kernel_launch(void* const* d_in, const int* in_sizes, int n_in,
                              void* d_out, int out_size, void* d_ws, size_t ws_size,
                              hipStream_t stream) {
    const float* x = (const float*)d_in[0];
    const float* s = (const float*)d_in[1];
    float* y = (float*)d_out;

    char* ws = (char*)d_ws;
    unsigned* flags = (unsigned*)(ws + 0);
    float2*   stats = (float2*)(ws + 64);
    const size_t stats_bytes = (size_t)R_ * NROW * sizeof(float2);      // 131200
    const size_t mask_off = (64 + stats_bytes + 255) & ~(size_t)255;
    unsigned char* mask = (unsigned char*)(ws + mask_off);              // 3.36 MB

    (void)hipMemsetAsync(flags, 0, R_ * sizeof(unsigned), stream);
    qact_init<<<NROW, TPB, 0, stream>>>(x, s, mask, stats, flags);
    for (int r = 0; r < R_; ++r)
        qact_step<<<NROW, TPB, 0, stream>>>(x, s, y, mask, stats, flags, r);
}